// RippleDecoder_13700945674249
// MI455X (gfx1250) — compile-verified
//
#include <hip/hip_runtime.h>
#include <hip/hip_bf16.h>

// ---------------------------------------------------------------------------
// RippleDecoder for MI455X (gfx1250): fp32 WMMA hypernetwork + LDS-broadcast
// ripple decoder with native v_sin_f32.
// ---------------------------------------------------------------------------

#define BSZ    16
#define IN_DIM 512
#define HID    1024
#define SEQ    4096
#define RH     32
#define MID    3104              // 2*32*32 + 33*32
#define OUTD   97
#define RUN    (4 * MID)         // 12416
#define WDIM   (RUN + OUTD + 4)  // 12517
#define WPAD   12520             // WDIM rounded up: rows of m are 16B-aligned
#define NEGS   0.01f

typedef float v2f __attribute__((ext_vector_type(2)));
typedef float v8f __attribute__((ext_vector_type(8)));

// ---------------------------------------------------------------------------
// GEMM: C(16 x N) = A(16 x K) @ W(K x N) + bias, optional leaky-relu.
// One wave32 per 16-column tile; fp32 WMMA 16x16x4 (ISA 7.12.2 layouts):
//   A frag:  VGPR0 = {K=k (lanes 0-15), K=k+2 (lanes 16-31)}, VGPR1 = K+1/K+3
//   B frag:  mirrored (row k striped across lanes within a VGPR)
//   C/D:     VGPR v -> row (v + 8*half), col = n0 + (lane&15)
// OOB lanes clamp their column: a D column depends only on the same B column,
// and clamped columns are never stored -> no branches in the K loop.
// Stores use row stride `ldc` (>= N) so the output rows can be kept aligned.
// ---------------------------------------------------------------------------
template <int ACT>
__global__ __launch_bounds__(32)
void gemm16_wmma(const float* __restrict__ A, const float* __restrict__ W,
                 const float* __restrict__ bias, float* __restrict__ C,
                 int K, int N, int ldc) {
    const int lane = threadIdx.x;
    const int half = lane >> 4;       // 0: lanes 0-15, 1: lanes 16-31
    const int lr   = lane & 15;
    const int col  = blockIdx.x * 16 + lr;
    const int colc = (col < N) ? col : (N - 1);

    v8f acc = {0.f, 0.f, 0.f, 0.f, 0.f, 0.f, 0.f, 0.f};

    const float*  aptr = A + (size_t)lr * K + 2 * half;
    const float*  bp0  = W + (size_t)(2 * half) * N + colc;  // row k+2h
    const float*  bp1  = bp0 + N;                            // row k+2h+1
    const size_t  step = (size_t)4 * N;

    #pragma unroll 4
    for (int k = 0; k < K; k += 4) {
        v2f a = *(const v2f*)aptr;           // A[lr][k+2h], A[lr][k+2h+1]
        v2f b = {*bp0, *bp1};
        acc = __builtin_amdgcn_wmma_f32_16x16x4_f32(
                  false, a, false, b, (short)0, acc, false, false);
        aptr += 4; bp0 += step; bp1 += step;
    }

    if (col < N) {
        const float bv = bias[col];
        #pragma unroll
        for (int v = 0; v < 8; ++v) {
            float r = acc[v] + bv;
            if (ACT) r = (r >= 0.f) ? r : NEGS * r;
            C[(size_t)(v + 8 * half) * ldc + col] = r;
        }
    }
}

// ---------------------------------------------------------------------------
// LayerNorm + leaky over rows of 1024. 16 blocks x 256 threads.
// ---------------------------------------------------------------------------
__global__ __launch_bounds__(256)
void ln_leaky_kernel(const float* __restrict__ tin, const float* __restrict__ g,
                     const float* __restrict__ be, float* __restrict__ out) {
    const int row = blockIdx.x;
    const int tid = threadIdx.x;
    const float* r = tin + (size_t)row * HID;

    float v[4], s = 0.f, ss = 0.f;
    #pragma unroll
    for (int k = 0; k < 4; ++k) {
        const float x = r[tid + 256 * k];
        v[k] = x; s += x; ss += x * x;
    }

    __shared__ float s1[256], s2[256];
    s1[tid] = s; s2[tid] = ss;
    __syncthreads();
    for (int off = 128; off > 0; off >>= 1) {
        if (tid < off) { s1[tid] += s1[tid + off]; s2[tid] += s2[tid + off]; }
        __syncthreads();
    }
    const float mean = s1[0] * (1.f / HID);
    const float var  = s2[0] * (1.f / HID) - mean * mean;
    const float inv  = rsqrtf(var + 1e-5f);

    #pragma unroll
    for (int k = 0; k < 4; ++k) {
        const int col = tid + 256 * k;
        float y = (v[k] - mean) * inv * g[col] + be[col];
        out[(size_t)row * HID + col] = (y >= 0.f) ? y : NEGS * y;
    }
}

// ---------------------------------------------------------------------------
// Ripple decoder: one thread per (b,t). Layer weights (3104 f32 = 12.4 KB)
// staged into LDS per layer via float4 (rows of m are 16B-aligned thanks to
// WPAD); all 256 threads broadcast-read the same LDS address each step
// (conflict-free). __sinf -> v_sin_f32 (TRANS, co-executes with the FMAs).
// ---------------------------------------------------------------------------
__global__ __launch_bounds__(256)
void ripple_kernel(const float* __restrict__ m, const float* __restrict__ Wfc,
                   const float* __restrict__ bfc, float* __restrict__ out) {
    const int b = blockIdx.y;
    const int t = blockIdx.x * 256 + threadIdx.x;
    const float* mb = m + (size_t)b * WPAD;

    __shared__ float sw[MID];

    // lc = coord @ Wfc + bfc ; coord = t * MULT (MULT = 1.0)
    const float coord = (float)t;
    float x[RH];
    #pragma unroll
    for (int j = 0; j < RH; ++j) x[j] = fmaf(coord, Wfc[j], bfc[j]);

    // 4 ripple layers 32 -> 32
    for (int l = 0; l < 4; ++l) {
        __syncthreads();
        {
            const float4* src = (const float4*)(mb + l * MID);
            float4*       dst = (float4*)sw;
            for (int idx = threadIdx.x; idx < MID / 4; idx += 256)
                dst[idx] = src[idx];
        }
        __syncthreads();

        float acc[RH];
        #pragma unroll
        for (int j = 0; j < RH; ++j) acc[j] = sw[3072 + j];  // bb[32][j]

        for (int i = 0; i < RH; ++i) {
            const float xi = x[i];
            #pragma unroll
            for (int j = 0; j < RH; ++j) {
                const v2f  wp = *(const v2f*)(sw + (i * RH + j) * 2); // w0,w1
                const float ph = sw[2048 + i * RH + j];               // bb[i][j]
                acc[j] = fmaf(wp.x, __sinf(fmaf(xi, wp.y, ph)), acc[j]);
            }
        }
        #pragma unroll
        for (int j = 0; j < RH; ++j) x[j] = acc[j];
    }

    // output layer 32 -> 1  (weights at mb[RUN .. RUN+96])
    __syncthreads();
    for (int idx = threadIdx.x; idx < OUTD; idx += 256)
        sw[idx] = mb[RUN + idx];
    __syncthreads();

    float v = sw[64 + 32];                 // bb[32]
    #pragma unroll
    for (int i = 0; i < RH; ++i)
        v = fmaf(sw[2 * i], __sinf(fmaf(x[i], sw[2 * i + 1], sw[64 + i])), v);

    // bypass 1 -> 1: reference slices byp_w from the SAME offset (run..run+4)
    // -> w0=sw[0], w1=sw[1], bb0=sw[2], bb1=sw[3]
    const float byp = fmaf(sw[0], __sinf(fmaf(v, sw[1], sw[2])), sw[3]);
    v = v + byp;

    out[(size_t)b * SEQ + t] = tanhf(v);
}

// ---------------------------------------------------------------------------
// Launcher
// ---------------------------------------------------------------------------
extern "C" void kernel_launch(void* const* d_in, const int* in_sizes, int n_in,
                              void* d_out, int out_size, void* d_ws, size_t ws_size,
                              hipStream_t stream) {
    (void)in_sizes; (void)n_in; (void)out_size; (void)ws_size;

    const float* x   = (const float*)d_in[0];
    const float* W0  = (const float*)d_in[1];
    const float* b0  = (const float*)d_in[2];
    const float* W1  = (const float*)d_in[3];
    const float* b1  = (const float*)d_in[4];
    const float* g1  = (const float*)d_in[5];
    const float* be1 = (const float*)d_in[6];
    const float* W2  = (const float*)d_in[7];
    const float* b2  = (const float*)d_in[8];
    const float* g2  = (const float*)d_in[9];
    const float* be2 = (const float*)d_in[10];
    const float* Wf  = (const float*)d_in[11];
    const float* bf  = (const float*)d_in[12];
    const float* Wfc = (const float*)d_in[13];
    const float* bfc = (const float*)d_in[14];

    float* ws  = (float*)d_ws;
    float* h0  = ws;                    // 16*1024
    float* t1  = ws + 1 * BSZ * HID;    // 16*1024
    float* h1  = ws + 2 * BSZ * HID;    // 16*1024
    float* mbf = ws + 3 * BSZ * HID;    // 16*12520 (padded rows, 16B-aligned)

    // h0 = leaky(x @ W0 + b0)
    gemm16_wmma<1><<<HID / 16, 32, 0, stream>>>(x, W0, b0, h0, IN_DIM, HID, HID);
    // h1 = leaky(ln(h0 @ W1 + b1))
    gemm16_wmma<0><<<HID / 16, 32, 0, stream>>>(h0, W1, b1, t1, HID, HID, HID);
    ln_leaky_kernel<<<BSZ, 256, 0, stream>>>(t1, g1, be1, h1);
    // h2 = leaky(ln(h1 @ W2 + b2))   (reuse h0 as h2)
    gemm16_wmma<0><<<HID / 16, 32, 0, stream>>>(h1, W2, b2, t1, HID, HID, HID);
    ln_leaky_kernel<<<BSZ, 256, 0, stream>>>(t1, g2, be2, h0);
    // m = h2 @ Wf + bf   (N = 12517 logical, rows stored with stride 12520)
    gemm16_wmma<0><<<(WDIM + 15) / 16, 32, 0, stream>>>(h0, Wf, bf, mbf,
                                                        HID, WDIM, WPAD);

    // ripple decoder + tanh + transpose
    dim3 grid(SEQ / 256, BSZ);
    ripple_kernel<<<grid, 256, 0, stream>>>(mbf, Wfc, bfc, (float*)d_out);
}